// Encoder_9552007266821
// MI455X (gfx1250) — compile-verified
//
#include <hip/hip_runtime.h>
#include <math.h>

// ---------------------------------------------------------------------------
// FISTA sparse coding, fused-resident CDNA5 (gfx1250) implementation.
//   A (161x161, padded 176x176 fp32) lives in LDS, per-wave y tile (176x32)
//   lives in LDS, x / DtY / accumulators live in VGPRs.
//   Matmul: V_WMMA_F32_16X16X4_F32, 11 M-tiles x 44 K-steps x 2 N-tiles
//   per iteration; A-operand loads amortized over both N-tiles so the loop
//   is XDL-bound rather than LDS-issue-bound.
// ---------------------------------------------------------------------------

#define T_FRAMES 36
#define NPOLES   40
#define KDIM     161          // 4*N+1
#define KPAD     176          // 11 * 16
#define MT       11           // KPAD/16  output row tiles
#define JT       44           // KPAD/4   reduction steps
#define NT       2            // N (column) tiles of 16 per wave
#define PCOLS    16384
#define MAXITER  100
#define LAMBD0   0.01f

#define WAVES    4            // waves per workgroup, NT*16 columns each
#define YSTRIDE  178          // column stride of y tile (even, bank-friendly)
#define DTSTRIDE 38           // column stride of Dt / Ystage tiles

// workspace layout (floats)
#define WS_A    0                       // KPAD*KPAD padded A = I - DtD/L
#define WS_DT   (KPAD*KPAD)             // KPAD*T_FRAMES padded Dt[k][t]
#define WS_SCAL (WS_DT + KPAD*T_FRAMES) // [0]=1/L  [1]=lambda/L

typedef __attribute__((ext_vector_type(2))) float v2f;
typedef __attribute__((ext_vector_type(8))) float v8f;

// ---------------------------------------------------------------------------
// Prep kernel (1 workgroup): build normalized dictionary, DtD, L, A, Dt.
// One-time ~2 MFLOP; performance irrelevant.
// ---------------------------------------------------------------------------
__global__ __launch_bounds__(256) void fista_prep(const float* __restrict__ Drr,
                                                  const float* __restrict__ Dtheta,
                                                  float* __restrict__ ws) {
    __shared__ float Dlds[T_FRAMES * KPAD];   // D[t][k], padded cols zero
    __shared__ float red[256];
    const int tid = threadIdx.x;

    for (int e = tid; e < T_FRAMES * KPAD; e += 256) Dlds[e] = 0.0f;
    __syncthreads();

    // raw dictionary: [1 | r^t cos | (-1)^t r^t cos | r^t sin | (-1)^t r^t sin]
    for (int e = tid; e < T_FRAMES * KDIM; e += 256) {
        int t = e / KDIM, k = e % KDIM;
        float v;
        if (k == 0) {
            v = 1.0f;
        } else {
            int grp = (k - 1) / NPOLES;
            int n   = (k - 1) % NPOLES;
            float powr = powf(Drr[n], (float)t);
            float sgn  = (t & 1) ? -1.0f : 1.0f;
            float s, c;
            sincosf((float)t * Dtheta[n], &s, &c);
            if      (grp == 0) v = powr * c;
            else if (grp == 1) v = sgn * powr * c;
            else if (grp == 2) v = powr * s;
            else               v = sgn * powr * s;
        }
        Dlds[t * KPAD + k] = v;
    }
    __syncthreads();

    // column normalization
    for (int k = tid; k < KDIM; k += 256) {
        float s = 0.0f;
        for (int t = 0; t < T_FRAMES; ++t) { float d = Dlds[t*KPAD + k]; s += d*d; }
        float g = sqrtf(s);
        if (g == 0.0f) g = sqrtf((float)T_FRAMES);
        float gi = 1.0f / g;
        for (int t = 0; t < T_FRAMES; ++t) Dlds[t*KPAD + k] *= gi;
    }
    __syncthreads();

    // pass 1: Frobenius norm of DtD
    float part = 0.0f;
    for (int e = tid; e < KDIM*KDIM; e += 256) {
        int k1 = e / KDIM, k2 = e % KDIM;
        float s = 0.0f;
        for (int t = 0; t < T_FRAMES; ++t) s += Dlds[t*KPAD + k1] * Dlds[t*KPAD + k2];
        part += s * s;
    }
    red[tid] = part;
    __syncthreads();
    for (int off = 128; off > 0; off >>= 1) {
        if (tid < off) red[tid] += red[tid + off];
        __syncthreads();
    }
    const float linv  = 1.0f / sqrtf(red[0]);
    const float lambd = LAMBD0 * linv;
    if (tid == 0) { ws[WS_SCAL] = linv; ws[WS_SCAL + 1] = lambd; }

    // pass 2: A = I - DtD/L, zero-padded to KPAD x KPAD
    for (int e = tid; e < KPAD*KPAD; e += 256) {
        int k1 = e / KPAD, k2 = e % KPAD;
        float a = 0.0f;
        if (k1 < KDIM && k2 < KDIM) {
            float s = 0.0f;
            for (int t = 0; t < T_FRAMES; ++t) s += Dlds[t*KPAD + k1] * Dlds[t*KPAD + k2];
            a = ((k1 == k2) ? 1.0f : 0.0f) - s * linv;
        }
        ws[WS_A + e] = a;
    }
    // Dt[k][t], zero rows k >= KDIM
    for (int e = tid; e < KPAD*T_FRAMES; e += 256) {
        int k = e / T_FRAMES, t = e % T_FRAMES;
        ws[WS_DT + e] = (k < KDIM) ? Dlds[t*KPAD + k] : 0.0f;
    }
}

// ---------------------------------------------------------------------------
// Main kernel: each wave runs full 100-iteration FISTA on 32 columns
// (two 16-wide N tiles). Grid covers all B*P = 65536 columns:
// 512 WGs x 4 waves x 32 cols.
// ---------------------------------------------------------------------------
__global__ __launch_bounds__(WAVES * 32) void fista_main(const float* __restrict__ Y,
                                                         const float* __restrict__ ws,
                                                         float* __restrict__ out) {
    __shared__ float Alds[KPAD * KPAD];                 // 123904 B
    __shared__ float Dtlds[KPAD * DTSTRIDE];            //  26752 B
    __shared__ float Ystg[WAVES][NT * 16 * DTSTRIDE];   //  19456 B
    __shared__ float Ytile[WAVES][NT * 16 * YSTRIDE];   //  91136 B

    const int tid   = threadIdx.x;
    const int wave  = tid >> 5;
    const int lane  = tid & 31;
    const int lm    = lane & 15;       // M row within tile / column id
    const int half2 = (lane >> 4) * 2; // K sub-offset for A/B operands
    const int half8 = (lane >> 4) * 8; // M sub-offset for C/D rows

    // cooperative load of A and Dt into LDS
    for (int e = tid; e < KPAD*KPAD; e += WAVES*32) Alds[e] = ws[WS_A + e];
    for (int e = tid; e < KPAD*T_FRAMES; e += WAVES*32) {
        int k = e / T_FRAMES, t = e % T_FRAMES;
        Dtlds[k * DTSTRIDE + t] = ws[WS_DT + e];
    }
    const float linv  = ws[WS_SCAL];
    const float lambd = ws[WS_SCAL + 1];

    // this wave's 32 columns (tiles never straddle a batch boundary:
    // PCOLS % (WAVES*NT*16) == 0)
    const int cg0   = blockIdx.x * (WAVES * NT * 16) + wave * (NT * 16);
    const int b     = cg0 / PCOLS;
    const int pbase = cg0 % PCOLS;

    // stage the (36 x 32) input tile, column-major
    for (int i = lane; i < T_FRAMES * NT * 16; i += 32) {
        int t = i >> 5, c = i & 31;
        Ystg[wave][c * DTSTRIDE + t] =
            Y[(size_t)b * T_FRAMES * PCOLS + (size_t)t * PCOLS + (pbase + c)];
    }
    // y starts at zero (including padded rows)
    for (int i = lane; i < NT * 16 * YSTRIDE; i += 32) Ytile[wave][i] = 0.0f;
    __syncthreads();

    float*       yt = &Ytile[wave][0];
    const float* ys = &Ystg[wave][0];

    const v8f zero8 = {0.f, 0.f, 0.f, 0.f, 0.f, 0.f, 0.f, 0.f};

    // ---- DtY = (1/L) * Dt @ Ytile_in, via WMMA (T=36 -> 9 K-steps) ----
    v8f dty[NT][MT];
#pragma unroll
    for (int nt = 0; nt < NT; ++nt)
#pragma unroll
        for (int mt = 0; mt < MT; ++mt) dty[nt][mt] = zero8;

    for (int j = 0; j < T_FRAMES / 4; ++j) {
        v2f bv0 = *(const v2f*)&ys[(lm     ) * DTSTRIDE + j * 4 + half2];
        v2f bv1 = *(const v2f*)&ys[(lm + 16) * DTSTRIDE + j * 4 + half2];
#pragma unroll
        for (int mt = 0; mt < MT; ++mt) {
            v2f av = *(const v2f*)&Dtlds[(mt * 16 + lm) * DTSTRIDE + j * 4 + half2];
            dty[0][mt] = __builtin_amdgcn_wmma_f32_16x16x4_f32(
                false, av, false, bv0, (short)0, dty[0][mt], false, false);
            dty[1][mt] = __builtin_amdgcn_wmma_f32_16x16x4_f32(
                false, av, false, bv1, (short)0, dty[1][mt], false, false);
        }
    }
#pragma unroll
    for (int nt = 0; nt < NT; ++nt)
#pragma unroll
        for (int mt = 0; mt < MT; ++mt)
#pragma unroll
            for (int r = 0; r < 8; ++r) dty[nt][mt][r] *= linv;

    // ---- FISTA iterations ----
    v8f xp[NT][MT];
#pragma unroll
    for (int nt = 0; nt < NT; ++nt)
#pragma unroll
        for (int mt = 0; mt < MT; ++mt) xp[nt][mt] = zero8;

    float tcur = 1.0f;
    for (int it = 0; it < MAXITER; ++it) {
        v8f acc[NT][MT];
#pragma unroll
        for (int nt = 0; nt < NT; ++nt)
#pragma unroll
            for (int mt = 0; mt < MT; ++mt) acc[nt][mt] = zero8;

        for (int j = 0; j < JT; ++j) {
            v2f bv0 = *(const v2f*)&yt[(lm     ) * YSTRIDE + j * 4 + half2];
            v2f bv1 = *(const v2f*)&yt[(lm + 16) * YSTRIDE + j * 4 + half2];
#pragma unroll
            for (int mt = 0; mt < MT; ++mt) {
                v2f av = *(const v2f*)&Alds[(mt * 16 + lm) * KPAD + j * 4 + half2];
                acc[0][mt] = __builtin_amdgcn_wmma_f32_16x16x4_f32(
                    false, av, false, bv0, (short)0, acc[0][mt], false, false);
                acc[1][mt] = __builtin_amdgcn_wmma_f32_16x16x4_f32(
                    false, av, false, bv1, (short)0, acc[1][mt], false, false);
            }
        }

        float tn = 0.5f * (1.0f + sqrtf(1.0f + 4.0f * tcur * tcur));
        float tt = (tcur - 1.0f) / tn;
        tcur = tn;

#pragma unroll
        for (int nt = 0; nt < NT; ++nt) {
#pragma unroll
            for (int mt = 0; mt < MT; ++mt) {
#pragma unroll
                for (int r = 0; r < 8; ++r) {
                    float z  = acc[nt][mt][r] + dty[nt][mt][r];
                    float az = fabsf(z) - lambd;
                    float xn = az > 0.0f ? copysignf(az, z) : 0.0f;
                    float yn = xn + tt * (xn - xp[nt][mt][r]);
                    xp[nt][mt][r] = xn;
                    yt[(lm + 16 * nt) * YSTRIDE + (mt * 16 + half8 + r)] = yn;
                }
            }
        }
    }

    // ---- write sparse codes (rows < 161) ----
#pragma unroll
    for (int nt = 0; nt < NT; ++nt) {
#pragma unroll
        for (int mt = 0; mt < MT; ++mt) {
#pragma unroll
            for (int r = 0; r < 8; ++r) {
                int Mr = mt * 16 + half8 + r;
                if (Mr < KDIM)
                    out[(size_t)b * KDIM * PCOLS + (size_t)Mr * PCOLS +
                        (pbase + 16 * nt + lm)] = xp[nt][mt][r];
            }
        }
    }
}

// ---------------------------------------------------------------------------
extern "C" void kernel_launch(void* const* d_in, const int* in_sizes, int n_in,
                              void* d_out, int out_size, void* d_ws, size_t ws_size,
                              hipStream_t stream) {
    const float* x      = (const float*)d_in[0];   // (B=4, T=36, P=16384)
    const float* Drr    = (const float*)d_in[1];   // (40,)
    const float* Dtheta = (const float*)d_in[2];   // (40,)
    float* out = (float*)d_out;                    // (4, 161, 16384)
    float* ws  = (float*)d_ws;

    fista_prep<<<1, 256, 0, stream>>>(Drr, Dtheta, ws);

    const int total_cols = 4 * PCOLS;                   // 65536
    const int wgs = total_cols / (WAVES * NT * 16);     // 512
    fista_main<<<wgs, WAVES * 32, 0, stream>>>(x, ws, out);
}